// STFT_7653631721942
// MI455X (gfx1250) — compile-verified
//
#include <hip/hip_runtime.h>
#include <math.h>

// ---------------- problem constants ----------------
#define FL     1024
#define HOP    256
#define CUTOFF 513
#define C2     1026      // 2*CUTOFF (basis rows / contraction dim of stage 3)
#define CP     1040      // C2 padded to multiple of 16
#define B_     8
#define T_     524288
#define NF     2049      // number of frames
#define NFP    2064      // NF padded to multiple of 16
#define NPAD   525312    // FL + HOP*(NF-1)
#define PAD    512

typedef __attribute__((ext_vector_type(2))) float v2f;
typedef __attribute__((ext_vector_type(8))) float v8f;

// D(16x16,f32) = A(16x4,f32) x B(4x16,f32) + C   -- full fp32 matrix op
#define WMMA_F32_4(a, b, c) \
  __builtin_amdgcn_wmma_f32_16x16x4_f32(false, (a), false, (b), (short)0, (c), false, false)

// ---------------- prep: pack fb -> fbP[k2][c] = {fb[c][2k2], fb[c][2k2+1]} ----------------
// One b64 load per B-fragment in k_fwd; c-padding pre-zeroed.
__global__ void k_prep_fbP(const float* __restrict__ fb, float2* __restrict__ fbP) {
  int idx = blockIdx.x * 256 + threadIdx.x;       // over (FL/2)*CP
  if (idx >= (FL / 2) * CP) return;
  int k2 = idx / CP, c = idx % CP;
  float2 v = {0.f, 0.f};
  if (c < C2) { v.x = fb[c * FL + 2 * k2]; v.y = fb[c * FL + 2 * k2 + 1]; }
  fbP[idx] = v;
}

// ---------------- prep: pack inv -> invP[c2][k] = {inv[2c2][k], inv[2c2+1][k]} ----------------
__global__ void k_prep_invP(const float* __restrict__ invb, float2* __restrict__ invP) {
  int idx = blockIdx.x * 256 + threadIdx.x;       // over (CP/2)*FL
  if (idx >= (CP / 2) * FL) return;
  int c2 = idx / FL, k = idx % FL;
  float2 v = {0.f, 0.f};
  if (2 * c2     < C2) v.x = invb[(2 * c2)     * FL + k];
  if (2 * c2 + 1 < C2) v.y = invb[(2 * c2 + 1) * FL + k];
  invP[idx] = v;
}

__global__ void k_zero(float* __restrict__ p, int n) {
  int i = blockIdx.x * 256 + threadIdx.x;
  if (i < n) p[i] = 0.f;
}

// ---------------- stage 1: ft[b][t][c] = sum_k frame(b,t,k) * fb[c][k] ----------------
// grid: x = c-tile group (9), y = t-tile (129), z = batch (8); 256 thr = 8 waves
__global__ __launch_bounds__(256) void k_fwd(const float* __restrict__ x,
                                             const float2* __restrict__ fbP,
                                             float* __restrict__ ftb) {
  // swizzled window: logical sample i stored at i + 4*(i>>8)  (frame rows 260 apart,
  // 260 mod 64 = 4 -> conflict-free b64 reads; even -> 8B aligned)
  __shared__ float xs[4936];
  const int t0 = blockIdx.y * 16;
  const int b  = blockIdx.z;
  const long g0 = (long)t0 * HOP;

  for (int i = threadIdx.x; i < 4864; i += 256) {
    long src = g0 + i - PAD;                      // reflect padding
    if (src < 0) src = -src;
    if (src >= T_) src = 2L * T_ - 2 - src;       // stays in [0,T) for all tiles
    xs[i + 4 * (i >> 8)] = x[(long)b * T_ + src];
  }
  __syncthreads();

  const int wave = threadIdx.x >> 5, lane = threadIdx.x & 31;
  const int ctile = blockIdx.x * 8 + wave;
  if (ctile >= CP / 16) return;                   // uniform per-wave exit
  const int lhi = lane >> 4;                      // lane-half selects K pair
  const int mr  = lane & 15;                      // A row (t) / B col (c)
  const int nc  = ctile * 16 + mr;

  v8f acc = {0.f, 0.f, 0.f, 0.f, 0.f, 0.f, 0.f, 0.f};
  #pragma unroll
  for (int kseg = 0; kseg < 4; ++kseg) {          // swizzle is linear within a segment
    const float*  xsb = &xs[(mr + kseg) * 260 + 2 * lhi];
    const float2* fbb = &fbP[(size_t)(kseg * 128 + lhi) * CP + nc];
    #pragma unroll 4
    for (int kc = 0; kc < 256; kc += 4) {
      const float2 av = *(const float2*)&xsb[kc]; // ds_load_b64, immediate offset
      const float2 bv = fbb[(size_t)(kc >> 1) * CP]; // global_load_b64, imm offset
      v2f a;  a.x  = av.x; a.y  = av.y;
      v2f bq; bq.x = bv.x; bq.y = bv.y;
      acc = WMMA_F32_4(a, bq, acc);
    }
  }

  // Unconditional stores: k_recomb deterministically rewrites pad rows/cols.
  float* dst = ftb + ((long)b * NFP + (t0 + lhi * 8)) * CP + nc;
  #pragma unroll
  for (int vi = 0; vi < 8; ++vi) dst[(size_t)vi * CP] = acc[vi];
}

// ---------------- stage 2: mag/phase recombination (reference-exact) ----------------
__global__ void k_recomb(float* __restrict__ ft) {
  const long idx = (long)blockIdx.x * 256 + threadIdx.x;  // over B_*NFP*CP
  if (idx >= (long)B_ * NFP * CP) return;
  const int c = (int)(idx % CP);
  const int t = (int)((idx / CP) % NFP);
  if (t >= NF || c >= C2) { ft[idx] = 0.f; return; }      // zero pad rows/cols
  if (c < CUTOFF) {
    const float r = ft[idx], im = ft[idx + CUTOFF];
    const float m  = sqrtf(r * r + im * im);
    const float ph = atan2f(im, r);
    ft[idx]          = m * cosf(ph);
    ft[idx + CUTOFF] = m * sinf(ph);
  }
}

// ---------------- stage 3: contrib = recomb @ inv, overlap-add via atomics ----------------
// grid: x = k-tile group (8), y = t-tile (129), z = batch (8); 256 thr = 8 waves
__global__ __launch_bounds__(256) void k_inv(const float* __restrict__ rc,
                                             const float2* __restrict__ invP,
                                             float* __restrict__ outp) {
  __shared__ float As[16 * 82];                   // 16 t-rows x 80 c-chunk, stride 82
  const int t0 = blockIdx.y * 16;
  const int b  = blockIdx.z;
  const int wave = threadIdx.x >> 5, lane = threadIdx.x & 31;
  const int k0   = (blockIdx.x * 8 + wave) * 16;  // 64 k-tiles, all valid
  const int lhi  = lane >> 4;
  const int mr   = lane & 15;
  const int kcol = k0 + mr;
  const long abase  = ((long)b * NFP + t0) * CP;
  const int  arow_l = mr * 82 + 2 * lhi;

  v8f acc = {0.f, 0.f, 0.f, 0.f, 0.f, 0.f, 0.f, 0.f};
  for (int cc0 = 0; cc0 < CP; cc0 += 80) {        // 13 chunks
    __syncthreads();
    for (int i = threadIdx.x; i < 16 * 80; i += 256) {
      const int tr = i / 80, cc = i - tr * 80;    // A tile shared by all 8 waves
      As[tr * 82 + cc] = rc[abase + (long)tr * CP + cc0 + cc];
    }
    __syncthreads();
    const float2* ibb = &invP[(size_t)((cc0 >> 1) + lhi) * FL + kcol];
    #pragma unroll 4
    for (int kc = 0; kc < 80; kc += 4) {
      const float2 av = *(const float2*)&As[arow_l + kc];  // ds_load_b64
      const float2 bv = ibb[(size_t)(kc >> 1) * FL];       // global_load_b64
      v2f a;  a.x  = av.x; a.y  = av.y;
      v2f bq; bq.x = bv.x; bq.y = bv.y;
      acc = WMMA_F32_4(a, bq, acc);
    }
  }

  const long ob = (long)b * NPAD + (long)t0 * HOP + kcol;
  if (t0 + 16 <= NF) {                            // fast path: no masking needed
    #pragma unroll
    for (int vi = 0; vi < 8; ++vi)
      atomicAdd(&outp[ob + (lhi * 8 + vi) * HOP], acc[vi]);
  } else {                                        // only last t-tile takes this path
    #pragma unroll
    for (int vi = 0; vi < 8; ++vi) {
      const int t = t0 + lhi * 8 + vi;
      if (t < NF) atomicAdd(&outp[(long)b * NPAD + t * HOP + kcol], acc[vi]);
    }
  }
}

// ---------------- stage 4: window-energy normalization, x4, crop ----------------
__global__ void k_norm(const float* __restrict__ outp, float* __restrict__ out) {
  const long idx = (long)blockIdx.x * 256 + threadIdx.x;  // B_*T_
  if (idx >= (long)B_ * T_) return;
  const int sout = (int)(idx % T_);
  const int b    = (int)(idx / T_);
  const int s    = sout + PAD;
  float wss = 0.f;
  const int tmax = s >> 8;
  #pragma unroll
  for (int d = 0; d < 4; ++d) {
    const int t = tmax - d;
    if (t >= 0 && t < NF) {
      const int j = s - (t << 8);
      if (j < FL) {
        const float w = 0.5f - 0.5f * cosf((float)j * 6.135923151e-3f); // 2*pi/1024
        wss += w * w;
      }
    }
  }
  float v = outp[(long)b * NPAD + s];
  if (wss > 1.17549435e-38f) v /= wss;            // np.finfo(f32).tiny
  out[idx] = v * 4.0f;                            // * FL/HOP
}

// ---------------- launcher ----------------
extern "C" void kernel_launch(void* const* d_in, const int* in_sizes, int n_in,
                              void* d_out, int out_size, void* d_ws, size_t ws_size,
                              hipStream_t stream) {
  const float* x    = (const float*)d_in[0];   // (8, 524288)
  const float* fb   = (const float*)d_in[1];   // (1026, 1, 1024)
  const float* invb = (const float*)d_in[2];   // (1026, 1, 1024)
  float* out = (float*)d_out;                  // (8, 1, 524288)

  char* ws = (char*)d_ws;                      // needs ~94 MB
  float2* fbP  = (float2*)ws;                                            // (FL/2)*CP
  float2* invP = (float2*)(ws + (size_t)(FL / 2) * CP * 8);              // (CP/2)*FL
  float*  ftb  = (float*)(ws + (size_t)(FL / 2) * CP * 8
                             + (size_t)(CP / 2) * FL * 8);               // B_*NFP*CP
  float*  outp = (float*)(ws + (size_t)(FL / 2) * CP * 8
                             + (size_t)(CP / 2) * FL * 8
                             + (size_t)B_ * NFP * CP * 4);               // B_*NPAD

  k_prep_fbP <<<((FL / 2) * CP + 255) / 256, 256, 0, stream>>>(fb, fbP);
  k_prep_invP<<<((CP / 2) * FL + 255) / 256, 256, 0, stream>>>(invb, invP);
  k_zero<<<(B_ * NPAD + 255) / 256, 256, 0, stream>>>(outp, B_ * NPAD);

  dim3 g1((CP / 16 + 7) / 8, NFP / 16, B_);    // (9, 129, 8)
  k_fwd<<<g1, 256, 0, stream>>>(x, fbP, ftb);

  const long tot = (long)B_ * NFP * CP;
  k_recomb<<<(int)((tot + 255) / 256), 256, 0, stream>>>(ftb);

  dim3 g2(8, NFP / 16, B_);                    // (8, 129, 8) -> 64 k-tiles x 8 waves
  k_inv<<<g2, 256, 0, stream>>>(ftb, invP, outp);

  k_norm<<<(int)(((long)B_ * T_ + 255) / 256), 256, 0, stream>>>(outp, out);
}